// SpatialAttentionBlock_8890582302951
// MI455X (gfx1250) — compile-verified
//
#include <hip/hip_runtime.h>

typedef float v4f __attribute__((ext_vector_type(4)));
typedef float v2f __attribute__((ext_vector_type(2)));
typedef float v8f __attribute__((ext_vector_type(8)));

#define B_   16
#define C_   256
#define H_   128
#define W_   128
#define HW_  (H_ * W_)    // 16384
#define HW4_ (HW_ / 4)    // 4096
#define CSPLIT_ 2
#define CCHUNK_ (C_ / CSPLIT_)   // 128

// ---------------------------------------------------------------------------
// Pass 1a: partial channel pool (max + sum over 128 channels per thread).
// 131072 threads = 4096 waves, unroll 16 -> ~32 MB of NT b128 loads in
// flight, enough to cover HBM latency at 23.3 TB/s.
// ---------------------------------------------------------------------------
__global__ void pool_partial_kernel(const float* __restrict__ x,
                                    v4f* __restrict__ pmax,
                                    v4f* __restrict__ psum) {
    int tid = blockIdx.x * blockDim.x + threadIdx.x;   // 0 .. 131071
    int s   = tid >> 16;                               // channel-split id 0/1
    int col = tid & 65535;                             // (b, q) column
    int b   = col >> 12;
    int q   = col & (HW4_ - 1);

    const v4f* xp = (const v4f*)x + (size_t)b * C_ * HW4_
                    + (size_t)s * CCHUNK_ * HW4_ + q;

    v4f v   = __builtin_nontemporal_load(xp);
    v4f mx  = v;
    v4f sum = v;
#pragma unroll 16
    for (int c = 1; c < CCHUNK_; ++c) {
        v4f t = __builtin_nontemporal_load(xp + (size_t)c * HW4_);
        mx.x = fmaxf(mx.x, t.x);
        mx.y = fmaxf(mx.y, t.y);
        mx.z = fmaxf(mx.z, t.z);
        mx.w = fmaxf(mx.w, t.w);
        sum += t;
    }
    pmax[tid] = mx;   // [CSPLIT][B*HW4]
    psum[tid] = sum;
}

// ---------------------------------------------------------------------------
// Pass 1b: combine the two partials into final pool [B, 2, H, W].
// ---------------------------------------------------------------------------
__global__ void pool_combine_kernel(const v4f* __restrict__ pmax,
                                    const v4f* __restrict__ psum,
                                    v4f* __restrict__ pool) {
    int tid = blockIdx.x * blockDim.x + threadIdx.x;   // 0 .. 65535
    int b = tid >> 12;
    int q = tid & (HW4_ - 1);

    v4f m0 = pmax[tid];
    v4f m1 = pmax[65536 + tid];
    v4f mx;
    mx.x = fmaxf(m0.x, m1.x);
    mx.y = fmaxf(m0.y, m1.y);
    mx.z = fmaxf(m0.z, m1.z);
    mx.w = fmaxf(m0.w, m1.w);
    v4f sum = psum[tid] + psum[65536 + tid];

    pool[(size_t)(b * 2 + 0) * HW4_ + q] = mx;
    pool[(size_t)(b * 2 + 1) * HW4_ + q] = sum * (1.0f / (float)C_);
}

// ---------------------------------------------------------------------------
// Pass 2: 7x7 conv (2 in-ch) + bias + BN + sigmoid via V_WMMA_F32_16X16X4_F32.
// Per wave: 16 consecutive-w pixels. A = 16x4 im2col taps (fp32), B = 4 taps
// broadcast to all 16 columns; 25 accumulating WMMAs cover K=98 (padded 100).
// Tap decode uses an LDS lookup table (packed {inv|ss|rr|delta}) built once
// per block, replacing the per-iteration divide-by-7/49 VALU chains.
// ---------------------------------------------------------------------------
__device__ __forceinline__ float tapfetch(const float* __restrict__ pb,
                                          int base, int h, int w, int pk) {
    int rr = (pk >> 16) & 0xff;
    int ss = (pk >> 24) & 0x7f;
    int hh = h + rr - 3;
    int ww = w + ss - 3;
    bool ok = (pk >= 0) & ((unsigned)hh < (unsigned)H_)
                        & ((unsigned)ww < (unsigned)W_);
    float v = 0.0f;
    if (ok) v = pb[base + (pk & 0xffff)];
    return v;
}

__global__ void conv_att_kernel(const float* __restrict__ pool,
                                const float* __restrict__ conv_w,
                                const float* __restrict__ conv_b,
                                const float* __restrict__ bn_gamma,
                                const float* __restrict__ bn_beta,
                                const float* __restrict__ bn_mean,
                                const float* __restrict__ bn_var,
                                float* __restrict__ att) {
    __shared__ float wlds[100];
    __shared__ int   tlds[100];
    int tix = threadIdx.x;
    if (tix < 100) {
        float wv = 0.0f;
        int   pk = (int)0x80000000;            // sign bit = invalid tap
        if (tix < 98) {
            wv = conv_w[tix];
            int ci = (tix >= 49) ? 1 : 0;
            int u  = tix - 49 * ci;
            int rr = u / 7;
            int ss = u - rr * 7;
            int delta = ci * HW_ + rr * W_ + ss;   // <= 17158, fits 16 bits
            pk = delta | (rr << 16) | (ss << 24);
        }
        wlds[tix] = wv;
        tlds[tix] = pk;
    }
    __syncthreads();

    int wave = tix >> 5;
    int lane = tix & 31;
    int g    = blockIdx.x * 8 + wave;   // 16-pixel group id, 0..16383 (exact)
    int p0   = g << 4;                  // flat pixel base in [B,H,W]
    int b    = p0 >> 14;
    int hw   = p0 & (HW_ - 1);
    int h    = hw >> 7;
    int w0   = hw & (W_ - 1);           // multiple of 16: never splits a row

    int m    = lane & 15;               // A-matrix row (pixel in group)
    int half = lane >> 4;               // K half per ISA 16x4 fp32 layout
    int w    = w0 + m;
    int base = (h - 3) * W_ + (w - 3);  // folded (h-3,w-3) origin

    const float* pb = pool + (size_t)b * 2 * HW_;

    v8f acc = {};
#pragma unroll 5
    for (int c = 0; c < 25; ++c) {
        int t0  = 4 * c + 2 * half;
        int pk0 = tlds[t0];
        int pk1 = tlds[t0 + 1];
        v2f a;
        a.x = tapfetch(pb, base, h, w, pk0);
        a.y = tapfetch(pb, base, h, w, pk1);
        v2f bw;
        bw.x = wlds[t0];
        bw.y = wlds[t0 + 1];
        // D = A(16x4) * B(4x16) + C ; fp32 exact accumulation
        acc = __builtin_amdgcn_wmma_f32_16x16x4_f32(
            false, a, false, bw, (short)0, acc, false, false);
    }

    float bias  = conv_b[0];
    float scale = bn_gamma[0] * rsqrtf(bn_var[0] + 1e-5f);
    float mean  = bn_mean[0];
    float beta  = bn_beta[0];

    if (m == 0) {                       // lanes 0 (rows 0-7) and 16 (rows 8-15)
        int obase = p0 + half * 8;
#pragma unroll
        for (int r = 0; r < 8; ++r) {
            float y = acc[r] + bias;
            y = (y - mean) * scale + beta;
            att[obase + r] = 1.0f / (1.0f + __expf(-y));
        }
    }
}

// ---------------------------------------------------------------------------
// Pass 3: out = x * att (broadcast over C). 268 MB NT in + 268 MB NT out;
// att (1 MB) stays cache-resident (regular temporal loads).
// ---------------------------------------------------------------------------
__global__ void scale_kernel(const float* __restrict__ x,
                             const float* __restrict__ att,
                             float* __restrict__ out) {
    size_t tid = (size_t)blockIdx.x * blockDim.x + threadIdx.x; // < 16777216
    const v4f* x4 = (const v4f*)x;
    const v4f* a4 = (const v4f*)att;
    v4f*       o4 = (v4f*)out;

    v4f v = __builtin_nontemporal_load(x4 + tid);
    size_t b   = tid >> 20;            // / (C_ * HW4_)
    size_t hw4 = tid & (HW4_ - 1);
    v4f a = a4[b * HW4_ + hw4];
    v4f r = v * a;
    __builtin_nontemporal_store(r, o4 + tid);
}

// ---------------------------------------------------------------------------
extern "C" void kernel_launch(void* const* d_in, const int* in_sizes, int n_in,
                              void* d_out, int out_size, void* d_ws, size_t ws_size,
                              hipStream_t stream) {
    (void)in_sizes; (void)n_in; (void)out_size; (void)ws_size;

    const float* x        = (const float*)d_in[0];
    const float* conv_w   = (const float*)d_in[1];
    const float* conv_b   = (const float*)d_in[2];
    const float* bn_gamma = (const float*)d_in[3];
    const float* bn_beta  = (const float*)d_in[4];
    const float* bn_mean  = (const float*)d_in[5];
    const float* bn_var   = (const float*)d_in[6];

    // Workspace layout (floats): pmax 2MB | psum 2MB | pool 2MB | att 1MB
    v4f*   pmax = (v4f*)d_ws;                          // [2][B*HW4] v4
    v4f*   psum = pmax + (size_t)CSPLIT_ * B_ * HW4_;  // [2][B*HW4] v4
    v4f*   pool4 = psum + (size_t)CSPLIT_ * B_ * HW4_; // [B,2,HW4] v4
    float* pool = (float*)pool4;
    float* att  = pool + (size_t)B_ * 2 * HW_;         // [B,H,W]

    // Pass 1a: 131072 threads, 128-channel partial reduction each.
    pool_partial_kernel<<<512, 256, 0, stream>>>(x, pmax, psum);

    // Pass 1b: 65536 threads, combine partials.
    pool_combine_kernel<<<256, 256, 0, stream>>>(pmax, psum, pool4);

    // Pass 2: 16384 wave-groups of 16 pixels, 8 waves per 256-thread block.
    conv_att_kernel<<<2048, 256, 0, stream>>>(pool, conv_w, conv_b,
                                              bn_gamma, bn_beta, bn_mean, bn_var,
                                              att);

    // Pass 3: 16,777,216 float4 threads.
    scale_kernel<<<65536, 256, 0, stream>>>(x, att, (float*)d_out);
}